// Attender_20263655702790
// MI455X (gfx1250) — compile-verified
//
#include <hip/hip_runtime.h>
#include <hip/hip_bf16.h>
#include <math.h>

typedef __attribute__((ext_vector_type(2))) float v2f;
typedef __attribute__((ext_vector_type(8))) float v8f;

#define NN 2
#define BB 32
#define KK 13      // landmarks
#define KP 16      // padded landmarks (WMMA N dim)
#define CC 512
#define HW 784     // 28*28

// ---------------------------------------------------------------------------
// Phase 1: softmax over the K(=13) axis at each (n,b,hw). Output is stored in
// an hw-paired, zero-padded layout:
//     recalT2[n][b][hw>>1][k(16)][hw&1]
// so that in phase 2 a lane's B-matrix fragment (elements (hw,k),(hw+1,k))
// is one contiguous 8-byte load.
// ---------------------------------------------------------------------------
__global__ __launch_bounds__(256) void softmaxT_kernel(
    const float* __restrict__ local_lm, float* __restrict__ recalT2) {
  int t  = blockIdx.x * blockDim.x + threadIdx.x;   // grid sized exactly
  int hw = t % HW;
  int nb = t / HW;                                   // n*BB + b
  const float* src = local_lm + ((size_t)nb * KK) * HW + hw;
  float x[KK];
  float m = -INFINITY;
#pragma unroll
  for (int k = 0; k < KK; ++k) { x[k] = src[(size_t)k * HW]; m = fmaxf(m, x[k]); }
  float s = 0.f;
#pragma unroll
  for (int k = 0; k < KK; ++k) { x[k] = __expf(x[k] - m); s += x[k]; }
  float inv = 1.0f / s;
  // paired layout: base of this hw's pair-group
  float* dst = recalT2 + (((size_t)nb * (HW / 2) + (hw >> 1)) * KP) * 2 + (hw & 1);
#pragma unroll
  for (int k = 0; k < KK; ++k) dst[2 * k] = x[k] * inv;
#pragma unroll
  for (int k = KK; k < KP; ++k) dst[2 * k] = 0.f;
}

// ---------------------------------------------------------------------------
// Phase 2: pooled[b, k, c] = sum_{n,hw} recal[n,b,k,hw] * gmaps[n,b,c,hw]
// One wave (32 threads) per (b, 16-wide c tile). GEMM with M=c(16), N=k(16),
// K_red = n*hw (1568), chained V_WMMA_F32_16X16X4_F32, f32 accumulate.
// ISA layouts (CDNA5 ISA):
//   A 16x4 f32 : lanes 0-15 M=0..15 K={0,1}; lanes 16-31 M=0..15 K={2,3}
//   B 4x16 f32 : VGPR0 lanes0-15 K=0 / lanes16-31 K=2; VGPR1 K=1 / K=3; N=lane&15
//   D 16x16 f32: VGPR i -> M = i + 8*(lane>=16), N = lane&15
// ---------------------------------------------------------------------------
__global__ __launch_bounds__(32) void wmma_pool_kernel(
    const float* __restrict__ gmaps, const float* __restrict__ recalT2,
    float* __restrict__ pooled /* [B,KK,CC] */) {
  const int b    = blockIdx.x;
  const int c0   = blockIdx.y * 16;
  const int lane = threadIdx.x;
  const int row  = lane & 15;          // A: M index (c); B: N column (landmark)
  const int koff = (lane >> 4) * 2;    // K sub-offset for this half-wave

  v8f acc = {};
  for (int n = 0; n < NN; ++n) {
    const float* gA = gmaps   + (((size_t)n * BB + b) * CC + (c0 + row)) * HW;
    const float* rB = recalT2 + ((size_t)n * BB + b) * HW * KP;
    for (int hw = 0; hw < HW; hw += 4) {
      v2f a  = *(const v2f*)(gA + hw + koff);                          // 8B A frag
      v2f bm = *(const v2f*)(rB + ((size_t)((hw + koff) >> 1) * KP + row) * 2); // 8B B frag
      acc = __builtin_amdgcn_wmma_f32_16x16x4_f32(
          /*neg_a=*/false, a, /*neg_b=*/false, bm,
          /*c_mod=*/(short)0, acc, /*reuse_a=*/false, /*reuse_b=*/false);
    }
  }

  const int chalf = (lane >> 4) * 8;
#pragma unroll
  for (int i = 0; i < 8; ++i) {
    int k = row;                // landmark
    int c = c0 + chalf + i;
    if (k < KK) pooled[((size_t)b * KK + k) * CC + c] = acc[i];
  }
}

// ---------------------------------------------------------------------------
// Phase 3: in-place L2 normalization of each (b,k) row over C=512.
// ---------------------------------------------------------------------------
__global__ __launch_bounds__(256) void normalize_kernel(float* __restrict__ out) {
  float* rowp = out + (size_t)blockIdx.x * CC;
  __shared__ float red[8];
  float ss = 0.f;
  for (int c = threadIdx.x; c < CC; c += 256) { float v = rowp[c]; ss += v * v; }
#pragma unroll
  for (int off = 16; off > 0; off >>= 1) ss += __shfl_down(ss, off, 32);
  if ((threadIdx.x & 31) == 0) red[threadIdx.x >> 5] = ss;
  __syncthreads();
  if (threadIdx.x == 0) {
    float tot = 0.f;
#pragma unroll
    for (int w = 0; w < 8; ++w) tot += red[w];
    red[0] = 1.0f / fmaxf(sqrtf(tot), 1e-12f);
  }
  __syncthreads();
  float scale = red[0];
  for (int c = threadIdx.x; c < CC; c += 256) rowp[c] *= scale;
}

extern "C" void kernel_launch(void* const* d_in, const int* in_sizes, int n_in,
                              void* d_out, int out_size, void* d_ws, size_t ws_size,
                              hipStream_t stream) {
  const float* local_lm = (const float*)d_in[0];  // [2,32,13,28,28]
  const float* gmaps    = (const float*)d_in[1];  // [2,32,512,28,28]
  float* out            = (float*)d_out;          // [32,13,512]
  float* recalT2        = (float*)d_ws;           // [2,32,392,16,2] = 3.2 MB

  // Phase 1: 2*32*784 = 50176 positions -> exactly 196 blocks of 256
  softmaxT_kernel<<<(NN * BB * HW) / 256, 256, 0, stream>>>(local_lm, recalT2);

  // Phase 2: one wave per (b, c-tile); 32 x 32 waves
  wmma_pool_kernel<<<dim3(BB, CC / 16), 32, 0, stream>>>(gmaps, recalT2, out);

  // Phase 3: 32*13 = 416 rows
  normalize_kernel<<<BB * KK, 256, 0, stream>>>(out);
}